// SEModule_18081812316801
// MI455X (gfx1250) — compile-verified
//
#include <hip/hip_runtime.h>

typedef __attribute__((ext_vector_type(2))) float v2f;
typedef __attribute__((ext_vector_type(4))) float v4f;
typedef __attribute__((ext_vector_type(8))) float v8f;

#define BATCH 64
#define CH    256
#define HID   16
#define HW    3136   // 56*56
#define HW4   784    // HW/4 (float4 granules, divides evenly)

// ---------------------------------------------------------------------------
// Kernel 1: global average pool. One block per (b,c) channel plane (3136
// contiguous floats). float4 streaming loads -> wave32 shuffle reduce ->
// LDS cross-wave reduce. Regular temporal hint: keep x hot in the 192MB L2
// so the scale pass mostly hits.
// ---------------------------------------------------------------------------
__global__ __launch_bounds__(256) void se_pool(const float* __restrict__ x,
                                               float* __restrict__ s) {
    const int bc = blockIdx.x;
    const v4f* xp = (const v4f*)(x + (size_t)bc * HW);

    float acc = 0.f;
    for (int i = threadIdx.x; i < HW4; i += 256) {
        v4f v = xp[i];
        acc += v.x + v.y + v.z + v.w;
    }
    // wave32 tree reduce
    #pragma unroll
    for (int off = 16; off > 0; off >>= 1)
        acc += __shfl_down(acc, off, 32);

    __shared__ float red[8];
    const int lane = threadIdx.x & 31;
    const int wv   = threadIdx.x >> 5;
    if (lane == 0) red[wv] = acc;
    __syncthreads();
    if (threadIdx.x == 0) {
        float t = 0.f;
        #pragma unroll
        for (int w = 0; w < 8; ++w) t += red[w];
        s[bc] = t * (1.0f / (float)HW);
    }
}

// ---------------------------------------------------------------------------
// Kernel 2: the SE bottleneck MLP, done in FP32 WMMA (V_WMMA_F32_16X16X4_F32).
// Single workgroup, 128 threads = 4 waves; wave w owns M-rows [16w, 16w+16).
//   h[64,16]  = relu(s[64,256] @ w1[16,256]^T + b1)     (K=256 -> 64 wmma)
//   g[64,256] = sigmoid(h[64,16] @ w2[256,16]^T + b2)   (16 N-tiles x 4 wmma)
//
// Fragment layouts per CDNA5 ISA 7.12.2 (32-bit, 16x4 A / 4x16 B / 16x16 C):
//   A: lane L -> row m=L%16; VGPR0 = A[m][k0], VGPR1 = A[m][k0+1],
//      k0 = (L<16 ? 0 : 2) within the K=4 step.
//   B: lane L -> col n=L%16; VGPR0 = B[k0][n], VGPR1 = B[k0+1][n].
//   C/D: VGPR v, lanes 0-15 -> row v, lanes 16-31 -> row v+8; col = L%16.
// ---------------------------------------------------------------------------
__global__ __launch_bounds__(128) void se_mlp(const float* __restrict__ s,
                                              const float* __restrict__ w1,
                                              const float* __restrict__ b1,
                                              const float* __restrict__ w2,
                                              const float* __restrict__ b2,
                                              float* __restrict__ g) {
    __shared__ float h_sm[BATCH][HID + 1];   // +1 pad to break bank conflicts

    const int tid   = threadIdx.x;
    const int lane  = tid & 31;
    const int hf    = lane >> 4;     // 0: K offset 0, 1: K offset 2
    const int l16   = lane & 15;
    const int wv    = tid >> 5;      // M-tile index (0..3)
    const int mrow0 = wv * 16;

    // ---- matmul 1: h = relu(s @ w1^T + b1) ---------------------------------
    v8f acc = {};
    for (int kk = 0; kk < CH; kk += 4) {
        const int k = kk + hf * 2;
        // A(m,k..k+1) = s[row, k..k+1] — contiguous, 8B aligned (k even)
        v2f a = *(const v2f*)(s + (size_t)(mrow0 + l16) * CH + k);
        // B(k..k+1, n) = w1[n, k..k+1] — contiguous in w1's row
        v2f b = *(const v2f*)(w1 + (size_t)l16 * CH + k);
        acc = __builtin_amdgcn_wmma_f32_16x16x4_f32(false, a, false, b,
                                                    (short)0, acc, false, false);
    }
    const float bias1 = b1[l16];
    #pragma unroll
    for (int v = 0; v < 8; ++v) {
        float hv = acc[v] + bias1;
        hv = hv > 0.f ? hv : 0.f;                  // relu
        h_sm[mrow0 + hf * 8 + v][l16] = hv;
    }
    __syncthreads();

    // ---- matmul 2: g = sigmoid(h @ w2^T + b2) ------------------------------
    for (int nt = 0; nt < CH / 16; ++nt) {
        const int n = nt * 16 + l16;
        v8f acc2 = {};
        #pragma unroll
        for (int kk = 0; kk < HID; kk += 4) {
            const int k = kk + hf * 2;
            v2f a;
            a.x = h_sm[mrow0 + l16][k];
            a.y = h_sm[mrow0 + l16][k + 1];
            v2f b = *(const v2f*)(w2 + (size_t)n * HID + k);
            acc2 = __builtin_amdgcn_wmma_f32_16x16x4_f32(false, a, false, b,
                                                         (short)0, acc2, false, false);
        }
        const float bias2 = b2[n];
        #pragma unroll
        for (int v = 0; v < 8; ++v) {
            float gv = acc2[v] + bias2;
            gv = 1.0f / (1.0f + __expf(-gv));      // sigmoid
            g[(size_t)(mrow0 + hf * 8 + v) * CH + n] = gv;
        }
    }
}

// ---------------------------------------------------------------------------
// Kernel 3: out = x * g[b,c]. One block per (b,c) so g is a uniform scalar
// load. x is last-use -> non-temporal load; out is never re-read ->
// non-temporal store (don't evict what's left of x in L2).
// ---------------------------------------------------------------------------
__global__ __launch_bounds__(256) void se_scale(const float* __restrict__ x,
                                                const float* __restrict__ g,
                                                float* __restrict__ out) {
    const int bc = blockIdx.x;
    const float gv = g[bc];
    const v4f* xp = (const v4f*)(x + (size_t)bc * HW);
    v4f*       op = (v4f*)(out + (size_t)bc * HW);
    for (int i = threadIdx.x; i < HW4; i += 256) {
        v4f v = __builtin_nontemporal_load(xp + i);
        v = v * gv;
        __builtin_nontemporal_store(v, op + i);
    }
}

// ---------------------------------------------------------------------------
extern "C" void kernel_launch(void* const* d_in, const int* in_sizes, int n_in,
                              void* d_out, int out_size, void* d_ws, size_t ws_size,
                              hipStream_t stream) {
    const float* x  = (const float*)d_in[0];   // [64,256,56,56]
    const float* w1 = (const float*)d_in[1];   // [16,256]
    const float* b1 = (const float*)d_in[2];   // [16]
    const float* w2 = (const float*)d_in[3];   // [256,16]
    const float* b2 = (const float*)d_in[4];   // [256]
    float* out = (float*)d_out;

    float* s_ws = (float*)d_ws;                // 64*256 floats
    float* g_ws = s_ws + BATCH * CH;           // 64*256 floats

    se_pool <<<BATCH * CH, 256, 0, stream>>>(x, s_ws);
    se_mlp  <<<1, 128, 0, stream>>>(s_ws, w1, b1, w2, b2, g_ws);
    se_scale<<<BATCH * CH, 256, 0, stream>>>(x, g_ws, out);
}